// CRFdepthUpsampler_50775103373911
// MI455X (gfx1250) — compile-verified
//
#include <hip/hip_runtime.h>
#include <math.h>

#define B_    4
#define C_    3
#define L_    18
#define H_    480
#define W_    640
#define H0_   120
#define W0_   160
#define R_    15
#define NPIX_ (H_ * W_)
#define P_    (B_ * NPIX_)
#define GAMMA_ 0.05f
#define EPS_   0.01f

typedef __attribute__((ext_vector_type(2))) float v2f;
typedef __attribute__((ext_vector_type(8))) float v8f;

// ---------------------------------------------------------------------------
// init: zero the max-reduction scalar (ws is poisoned with 0xAA by harness)
// ---------------------------------------------------------------------------
__global__ void init_kernel(unsigned* mx) {
    if (threadIdx.x == 0 && blockIdx.x == 0) mx[0] = 0u;
}

// ---------------------------------------------------------------------------
// bilinear upsample 120x160 -> 480x640 (half-pixel centers, edge clamp)
// ---------------------------------------------------------------------------
__global__ void upsample_kernel(const float* __restrict__ disp, float* __restrict__ up) {
    int t = blockIdx.x * blockDim.x + threadIdx.x;
    if (t >= P_) return;
    int b = t / NPIX_;
    int pix = t - b * NPIX_;
    int y = pix / W_, x = pix - y * W_;
    float sy = (y + 0.5f) * 0.25f - 0.5f;
    float sx = (x + 0.5f) * 0.25f - 0.5f;
    int y0 = (int)floorf(sy), x0 = (int)floorf(sx);
    float fy = sy - (float)y0, fx = sx - (float)x0;
    int y0c = min(max(y0, 0), H0_ - 1), y1c = min(max(y0 + 1, 0), H0_ - 1);
    int x0c = min(max(x0, 0), W0_ - 1), x1c = min(max(x0 + 1, 0), W0_ - 1);
    const float* dp = disp + (size_t)b * H0_ * W0_;
    float v00 = dp[y0c * W0_ + x0c], v01 = dp[y0c * W0_ + x1c];
    float v10 = dp[y1c * W0_ + x0c], v11 = dp[y1c * W0_ + x1c];
    up[t] = (1.f - fy) * ((1.f - fx) * v00 + fx * v01) + fy * ((1.f - fx) * v10 + fx * v11);
}

// ---------------------------------------------------------------------------
// global max of up (values >= 0 -> float bits monotonic as uint)
// ---------------------------------------------------------------------------
__global__ void reduce_max_kernel(const float* __restrict__ up, unsigned* mx) {
    __shared__ float s[256];
    int t = blockIdx.x * 256 + threadIdx.x;
    float v = (t < P_) ? up[t] : 0.f;
    s[threadIdx.x] = v;
    __syncthreads();
    for (int o = 128; o > 0; o >>= 1) {
        if (threadIdx.x < o) s[threadIdx.x] = fmaxf(s[threadIdx.x], s[threadIdx.x + o]);
        __syncthreads();
    }
    if (threadIdx.x == 0) atomicMax(mx, __float_as_uint(s[0]));
}

// ---------------------------------------------------------------------------
// cu = confidence * unary ; q_logits init = cu
// ---------------------------------------------------------------------------
__global__ void unary_kernel(const float* __restrict__ up, const float* __restrict__ mxf,
                             float* __restrict__ cu, float* __restrict__ ql) {
    int t = blockIdx.x * blockDim.x + threadIdx.x;
    if (t >= P_) return;
    int b = t / NPIX_;
    int pix = t - b * NPIX_;
    float u = up[t];
    float conf = (u > 0.01f) ? 1.f : 0.f;
    float step = mxf[0] * (1.0f / 17.0f);
    size_t base = (size_t)b * L_ * NPIX_ + pix;
    #pragma unroll
    for (int l = 0; l < L_; ++l) {
        float d = u - (float)l * step;
        float v = conf * (-10.0f * sqrtf(d * d + GAMMA_ * GAMMA_));
        cu[base + (size_t)l * NPIX_] = v;
        ql[base + (size_t)l * NPIX_] = v;
    }
}

// ---------------------------------------------------------------------------
// softmax over 18 labels per pixel
// ---------------------------------------------------------------------------
__global__ void softmax_kernel(const float* __restrict__ ql, float* __restrict__ Q) {
    int t = blockIdx.x * blockDim.x + threadIdx.x;
    if (t >= P_) return;
    int b = t / NPIX_;
    int pix = t - b * NPIX_;
    size_t base = (size_t)b * L_ * NPIX_ + pix;
    float v[L_], mxv = -1e30f;
    #pragma unroll
    for (int l = 0; l < L_; ++l) { v[l] = ql[base + (size_t)l * NPIX_]; mxv = fmaxf(mxv, v[l]); }
    float sum = 0.f;
    #pragma unroll
    for (int l = 0; l < L_; ++l) { v[l] = __expf(v[l] - mxv); sum += v[l]; }
    float inv = 1.0f / sum;
    #pragma unroll
    for (int l = 0; l < L_; ++l) Q[base + (size_t)l * NPIX_] = v[l] * inv;
}

// ---------------------------------------------------------------------------
// horizontal truncated box sum, one block per row (640 threads)
// ---------------------------------------------------------------------------
__global__ void hbox_kernel(const float* __restrict__ in, float* __restrict__ out) {
    __shared__ float s[W_ + 2 * R_];
    int row = blockIdx.x;
    int tid = threadIdx.x;
    const float* rp = in + (size_t)row * W_;
    int g = tid - R_;
    s[tid] = (g >= 0 && g < W_) ? rp[g] : 0.f;
    if (tid < 2 * R_) {
        int g2 = tid + W_ - R_;
        s[W_ + tid] = (g2 < W_) ? rp[g2] : 0.f;
    }
    __syncthreads();
    float sum = 0.f;
    #pragma unroll
    for (int k = 0; k <= 2 * R_; ++k) sum += s[tid + k];
    out[(size_t)row * W_ + tid] = sum;
}

// horizontal box of I[c] * Q[m], rows ordered as [b][c][m][y]
__global__ void hbox_prod_kernel(const float* __restrict__ img, const float* __restrict__ Q,
                                 float* __restrict__ out) {
    __shared__ float s[W_ + 2 * R_];
    int row = blockIdx.x;
    int tid = threadIdx.x;
    int y = row % H_;
    int ci = row / H_;
    int m = ci % L_;
    int c = (ci / L_) % C_;
    int b = ci / (L_ * C_);
    const float* ip = img + ((size_t)(b * C_ + c) * H_ + y) * W_;
    const float* qp = Q + ((size_t)(b * L_ + m) * H_ + y) * W_;
    int g = tid - R_;
    s[tid] = (g >= 0 && g < W_) ? ip[g] * qp[g] : 0.f;
    if (tid < 2 * R_) {
        int g2 = tid + W_ - R_;
        s[W_ + tid] = (g2 < W_) ? ip[g2] * qp[g2] : 0.f;
    }
    __syncthreads();
    float sum = 0.f;
    #pragma unroll
    for (int k = 0; k <= 2 * R_; ++k) sum += s[tid + k];
    out[(size_t)row * W_ + tid] = sum;
}

// horizontal box of I[c]*I[d] for the 6 unique symmetric pairs, rows [b][pair][y]
__global__ void hbox_ii_kernel(const float* __restrict__ img, float* __restrict__ out) {
    __shared__ float s[W_ + 2 * R_];
    const int pc[6] = {0, 0, 0, 1, 1, 2};
    const int pd[6] = {0, 1, 2, 1, 2, 2};
    int row = blockIdx.x;
    int tid = threadIdx.x;
    int y = row % H_;
    int ci = row / H_;
    int pr = ci % 6;
    int b = ci / 6;
    const float* ip = img + ((size_t)(b * C_ + pc[pr]) * H_ + y) * W_;
    const float* jp = img + ((size_t)(b * C_ + pd[pr]) * H_ + y) * W_;
    int g = tid - R_;
    s[tid] = (g >= 0 && g < W_) ? ip[g] * jp[g] : 0.f;
    if (tid < 2 * R_) {
        int g2 = tid + W_ - R_;
        s[W_ + tid] = (g2 < W_) ? ip[g2] * jp[g2] : 0.f;
    }
    __syncthreads();
    float sum = 0.f;
    #pragma unroll
    for (int k = 0; k <= 2 * R_; ++k) sum += s[tid + k];
    out[(size_t)row * W_ + tid] = sum;
}

// ---------------------------------------------------------------------------
// vertical box sum with fused /N (analytic counts), sliding-sum in LDS.
// Tile staging uses CDNA5 async global->LDS copies (GLOBAL_LOAD_ASYNC_TO_LDS_B32,
// GVS mode: SGPR 64-bit base + per-lane 32-bit offset), tracked by ASYNCcnt.
// Out-of-range halo rows are zero-filled with ordinary LDS stores (disjoint
// addresses from the async targets, so DScnt/ASYNCcnt disorder is safe).
// grid (W/64, H/32, nch), block (64,4); each thread: 8 consecutive rows
// ---------------------------------------------------------------------------
__global__ void vbox_kernel(const float* __restrict__ in, float* __restrict__ out) {
    __shared__ float s[32 + 2 * R_][64];
    int tx = threadIdx.x;
    int ty = threadIdx.y;
    int x0 = blockIdx.x * 64;
    int y0 = blockIdx.y * 32;
    size_t cbase = (size_t)blockIdx.z * H_ * W_;
    unsigned long long sbase = (unsigned long long)(const void*)(in + cbase);
    for (int rr = ty; rr < 32 + 2 * R_; rr += 4) {
        int gy = y0 + rr - R_;
        if (gy >= 0 && gy < H_) {
            unsigned int voff = (unsigned int)(((unsigned)gy * W_ + (unsigned)(x0 + tx)) * 4u);
            unsigned int ldsa = (unsigned int)(unsigned long long)(&s[rr][tx]);
            asm volatile("global_load_async_to_lds_b32 %0, %1, %2"
                         :
                         : "v"(ldsa), "v"(voff), "s"(sbase)
                         : "memory");
        } else {
            s[rr][tx] = 0.f;
        }
    }
    asm volatile("s_wait_asynccnt 0x0" ::: "memory");
    __syncthreads();
    int gx = x0 + tx;
    float icx = 1.0f / (float)(min(gx + R_, W_ - 1) - max(gx - R_, 0) + 1);
    int oyb = ty * 8;
    float sum = 0.f;
    #pragma unroll
    for (int k = 0; k <= 2 * R_; ++k) sum += s[oyb + k][tx];
    #pragma unroll
    for (int j = 0; j < 8; ++j) {
        int gy = y0 + oyb + j;
        float icy = 1.0f / (float)(min(gy + R_, H_ - 1) - max(gy - R_, 0) + 1);
        out[cbase + (size_t)gy * W_ + gx] = sum * icx * icy;
        if (j < 7) sum += s[oyb + j + 1 + 2 * R_][tx] - s[oyb + j][tx];
    }
}

// ---------------------------------------------------------------------------
// per-pixel inverse of Sigma = var_I + eps*I (symmetric 3x3, 6 floats)
// mII holds box-means of I*I (6 pairs)
// ---------------------------------------------------------------------------
__global__ void invsigma_kernel(const float* __restrict__ mII, const float* __restrict__ mI,
                                float* __restrict__ iS) {
    int t = blockIdx.x * blockDim.x + threadIdx.x;
    if (t >= P_) return;
    int b = t / NPIX_;
    int pix = t - b * NPIX_;
    float mi0 = mI[((size_t)(b * 3 + 0)) * NPIX_ + pix];
    float mi1 = mI[((size_t)(b * 3 + 1)) * NPIX_ + pix];
    float mi2 = mI[((size_t)(b * 3 + 2)) * NPIX_ + pix];
    float c0 = mII[((size_t)(b * 6 + 0)) * NPIX_ + pix];
    float c1 = mII[((size_t)(b * 6 + 1)) * NPIX_ + pix];
    float c2 = mII[((size_t)(b * 6 + 2)) * NPIX_ + pix];
    float c3 = mII[((size_t)(b * 6 + 3)) * NPIX_ + pix];
    float c4 = mII[((size_t)(b * 6 + 4)) * NPIX_ + pix];
    float c5 = mII[((size_t)(b * 6 + 5)) * NPIX_ + pix];
    float s00 = c0 - mi0 * mi0 + EPS_;
    float s01 = c1 - mi0 * mi1;
    float s02 = c2 - mi0 * mi2;
    float s11 = c3 - mi1 * mi1 + EPS_;
    float s12 = c4 - mi1 * mi2;
    float s22 = c5 - mi2 * mi2 + EPS_;
    float i00 = s11 * s22 - s12 * s12;
    float i01 = s02 * s12 - s01 * s22;
    float i02 = s01 * s12 - s02 * s11;
    float i11 = s00 * s22 - s02 * s02;
    float i12 = s01 * s02 - s00 * s12;
    float i22 = s00 * s11 - s01 * s01;
    float det = s00 * i00 + s01 * i01 + s02 * i02;
    float id = 1.0f / det;
    iS[((size_t)(b * 6 + 0)) * NPIX_ + pix] = i00 * id;
    iS[((size_t)(b * 6 + 1)) * NPIX_ + pix] = i01 * id;
    iS[((size_t)(b * 6 + 2)) * NPIX_ + pix] = i02 * id;
    iS[((size_t)(b * 6 + 3)) * NPIX_ + pix] = i11 * id;
    iS[((size_t)(b * 6 + 4)) * NPIX_ + pix] = i12 * id;
    iS[((size_t)(b * 6 + 5)) * NPIX_ + pix] = i22 * id;
}

// ---------------------------------------------------------------------------
// a = invSigma * (corr_Ip - mI*mp)  (in-place over corr), b = mp - a.mI
// ---------------------------------------------------------------------------
__global__ void solve_kernel(const float* __restrict__ mp, const float* __restrict__ mI,
                             const float* __restrict__ iS, float* __restrict__ aB,
                             float* __restrict__ bB) {
    int t = blockIdx.x * blockDim.x + threadIdx.x;
    if (t >= P_) return;
    int b = t / NPIX_;
    int pix = t - b * NPIX_;
    float mi[3], is[6];
    #pragma unroll
    for (int c = 0; c < 3; ++c) mi[c] = mI[((size_t)(b * 3 + c)) * NPIX_ + pix];
    #pragma unroll
    for (int j = 0; j < 6; ++j) is[j] = iS[((size_t)(b * 6 + j)) * NPIX_ + pix];
    #pragma unroll
    for (int m = 0; m < L_; ++m) {
        float mpv = mp[((size_t)(b * L_ + m)) * NPIX_ + pix];
        size_t i0 = ((size_t)((b * 3 + 0) * L_ + m)) * NPIX_ + pix;
        size_t i1 = ((size_t)((b * 3 + 1) * L_ + m)) * NPIX_ + pix;
        size_t i2 = ((size_t)((b * 3 + 2) * L_ + m)) * NPIX_ + pix;
        float cv0 = aB[i0] - mi[0] * mpv;
        float cv1 = aB[i1] - mi[1] * mpv;
        float cv2 = aB[i2] - mi[2] * mpv;
        float a0 = is[0] * cv0 + is[1] * cv1 + is[2] * cv2;
        float a1 = is[1] * cv0 + is[3] * cv1 + is[4] * cv2;
        float a2 = is[2] * cv0 + is[4] * cv1 + is[5] * cv2;
        aB[i0] = a0;
        aB[i1] = a1;
        aB[i2] = a2;
        bB[((size_t)(b * L_ + m)) * NPIX_ + pix] = mpv - (a0 * mi[0] + a1 * mi[1] + a2 * mi[2]);
    }
}

// ---------------------------------------------------------------------------
// fused: Qf = ma.I + mb ; msg = mu @ Qf via V_WMMA_F32_16X16X4_F32 ;
//        q_logits = cu - msg
// mu is 18x18, zero-padded in LDS to 32x20 so all A-operand reads are
// unconditional (no exec save/restore around the WMMAs). K padded 18->20
// (5 chunks of 4), M padded 18->32 (two 16-row blocks): 10 WMMAs / 16 pixels.
// A layout (16x4 f32): lane L -> M = L&15 ; v.x = K+2*(L>>4), v.y = K+1+2*(L>>4)
// B layout (4x16 f32): lane L -> N = L&15 ; same K split per half-wave
// C/D layout: VGPR j -> row j (lanes 0-15) / row j+8 (lanes 16-31), N = L&15
// ---------------------------------------------------------------------------
__global__ void apply_mu_kernel(const float* __restrict__ img, const float* __restrict__ ma,
                                const float* __restrict__ mb, const float* __restrict__ cu,
                                float* __restrict__ ql, const float* __restrict__ mxf) {
    __shared__ float mus[32 * 20];   // zero-padded mu
    float step = mxf[0] * (1.0f / 17.0f);
    int tid = threadIdx.x;
    for (int i = tid; i < 32 * 20; i += blockDim.x) {
        int l = i / 20, m = i - l * 20;
        float d = (float)(l - m) * step;
        mus[i] = (l < L_ && m < L_) ? sqrtf(d * d + GAMMA_ * GAMMA_) : 0.f;
    }
    __syncthreads();

    int lane = tid & 31;
    int wave = tid >> 5;
    int p = lane & 15;
    int hi = lane >> 4;
    int nwaves = gridDim.x * (blockDim.x >> 5);
    int wid = blockIdx.x * (blockDim.x >> 5) + wave;
    const int ngroups = P_ / 16;

    for (int g = wid; g < ngroups; g += nwaves) {
        int pg = g * 16 + p;            // groups never straddle batches (NPIX%16==0)
        int b = pg / NPIX_;
        int pix = pg - b * NPIX_;

        float ic0 = img[((size_t)(b * 3 + 0)) * NPIX_ + pix];
        float ic1 = img[((size_t)(b * 3 + 1)) * NPIX_ + pix];
        float ic2 = img[((size_t)(b * 3 + 2)) * NPIX_ + pix];

        // each half-wave computes the Qf labels its B registers need
        // (branchless: clamp index, mask out K-padding columns)
        float qf[5][2];
        #pragma unroll
        for (int k = 0; k < 5; ++k) {
            #pragma unroll
            for (int j = 0; j < 2; ++j) {
                int m = 4 * k + 2 * hi + j;
                int mc = min(m, L_ - 1);
                float msk = (m < L_) ? 1.f : 0.f;
                float v = mb[((size_t)(b * L_ + mc)) * NPIX_ + pix];
                v += ma[((size_t)((b * 3 + 0) * L_ + mc)) * NPIX_ + pix] * ic0;
                v += ma[((size_t)((b * 3 + 1) * L_ + mc)) * NPIX_ + pix] * ic1;
                v += ma[((size_t)((b * 3 + 2) * L_ + mc)) * NPIX_ + pix] * ic2;
                qf[k][j] = v * msk;
            }
        }

        v8f c0 = {};
        v8f c1 = {};
        int M0 = lane & 15;
        int M1 = 16 + M0;
        #pragma unroll
        for (int k = 0; k < 5; ++k) {
            int K0 = 4 * k + 2 * hi;
            v2f bb;
            bb.x = qf[k][0];
            bb.y = qf[k][1];
            v2f a0, a1;
            a0.x = mus[M0 * 20 + K0];
            a0.y = mus[M0 * 20 + K0 + 1];
            a1.x = mus[M1 * 20 + K0];
            a1.y = mus[M1 * 20 + K0 + 1];
            c0 = __builtin_amdgcn_wmma_f32_16x16x4_f32(false, a0, false, bb, (short)0, c0,
                                                       false, false);
            c1 = __builtin_amdgcn_wmma_f32_16x16x4_f32(false, a1, false, bb, (short)0, c1,
                                                       false, false);
        }

        // lane holds msg[j + 8*hi][p] in c0[j] and msg[16 + j + 8*hi][p] in c1[j]
        size_t obase = (size_t)b * L_ * NPIX_ + pix;
        #pragma unroll
        for (int j = 0; j < 8; ++j) {
            int l0 = j + 8 * hi;
            ql[obase + (size_t)l0 * NPIX_] = cu[obase + (size_t)l0 * NPIX_] - c0[j];
            int l1 = 16 + j + 8 * hi;
            if (l1 < L_)
                ql[obase + (size_t)l1 * NPIX_] = cu[obase + (size_t)l1 * NPIX_] - c1[j];
        }
    }
}

// ---------------------------------------------------------------------------
// final: probs = softmax(q_logits); out = sum_l probs[l] * label[l]
// ---------------------------------------------------------------------------
__global__ void final_kernel(const float* __restrict__ ql, const float* __restrict__ mxf,
                             float* __restrict__ out) {
    int t = blockIdx.x * blockDim.x + threadIdx.x;
    if (t >= P_) return;
    int b = t / NPIX_;
    int pix = t - b * NPIX_;
    float step = mxf[0] * (1.0f / 17.0f);
    size_t base = (size_t)b * L_ * NPIX_ + pix;
    float v[L_], mxv = -1e30f;
    #pragma unroll
    for (int l = 0; l < L_; ++l) { v[l] = ql[base + (size_t)l * NPIX_]; mxv = fmaxf(mxv, v[l]); }
    float sum = 0.f, acc = 0.f;
    #pragma unroll
    for (int l = 0; l < L_; ++l) {
        float e = __expf(v[l] - mxv);
        sum += e;
        acc += e * ((float)l * step);
    }
    out[t] = acc / sum;
}

// ---------------------------------------------------------------------------
extern "C" void kernel_launch(void* const* d_in, const int* in_sizes, int n_in,
                              void* d_out, int out_size, void* d_ws, size_t ws_size,
                              hipStream_t stream) {
    const float* disp = (const float*)d_in[0];   // (4,1,120,160)
    const float* img  = (const float*)d_in[1];   // (4,3,480,640)
    float* out = (float*)d_out;                  // (4,1,480,640)

    float* ws = (float*)d_ws;
    const size_t P = (size_t)P_;
    float* mxf = ws;                 // [0..15] scalar area
    float* up  = ws + 16;            // 1*P
    float* cu  = up + P;             // 18*P
    float* ql  = cu + 18 * P;        // 18*P
    float* Qb  = ql + 18 * P;        // 18*P
    float* mp  = Qb + 18 * P;        // 18*P
    float* aB  = mp + 18 * P;        // 54*P  (corr_Ip -> a -> ma ; also mII pre-loop)
    float* bB  = aB + 54 * P;        // 18*P  (b -> mb)
    float* tmp = bB + 18 * P;        // 54*P  (hbox staging)
    float* mI  = tmp + 54 * P;       // 3*P
    float* iS  = mI + 3 * P;         // 6*P

    const int T = 256;
    const int GP = (P_ + T - 1) / T;         // 4800
    dim3 vblk(64, 4);

    // --- preamble -----------------------------------------------------------
    init_kernel<<<1, 1, 0, stream>>>((unsigned*)mxf);
    upsample_kernel<<<GP, T, 0, stream>>>(disp, up);
    reduce_max_kernel<<<GP, T, 0, stream>>>(up, (unsigned*)mxf);
    unary_kernel<<<GP, T, 0, stream>>>(up, mxf, cu, ql);

    // mean of I (3 ch per batch)
    hbox_kernel<<<B_ * C_ * H_, W_, 0, stream>>>(img, tmp);
    vbox_kernel<<<dim3(W_ / 64, H_ / 32, B_ * C_), vblk, 0, stream>>>(tmp, mI);
    // mean of I*I (6 pairs per batch) -> aB, then invert Sigma once per pixel
    hbox_ii_kernel<<<B_ * 6 * H_, W_, 0, stream>>>(img, tmp);
    vbox_kernel<<<dim3(W_ / 64, H_ / 32, B_ * 6), vblk, 0, stream>>>(tmp, aB);
    invsigma_kernel<<<GP, T, 0, stream>>>(aB, mI, iS);

    // --- CRF mean-field iterations -----------------------------------------
    for (int it = 0; it < 2; ++it) {
        softmax_kernel<<<GP, T, 0, stream>>>(ql, Qb);

        // mp = box(Q)/N
        hbox_kernel<<<B_ * L_ * H_, W_, 0, stream>>>(Qb, tmp);
        vbox_kernel<<<dim3(W_ / 64, H_ / 32, B_ * L_), vblk, 0, stream>>>(tmp, mp);

        // corr_Ip = box(I*Q)/N  -> aB
        hbox_prod_kernel<<<B_ * C_ * L_ * H_, W_, 0, stream>>>(img, Qb, tmp);
        vbox_kernel<<<dim3(W_ / 64, H_ / 32, B_ * C_ * L_), vblk, 0, stream>>>(tmp, aB);

        // a (in-place over corr), b
        solve_kernel<<<GP, T, 0, stream>>>(mp, mI, iS, aB, bB);

        // ma = box(a)/N (in-place), mb = box(b)/N (in-place)
        hbox_kernel<<<B_ * C_ * L_ * H_, W_, 0, stream>>>(aB, tmp);
        vbox_kernel<<<dim3(W_ / 64, H_ / 32, B_ * C_ * L_), vblk, 0, stream>>>(tmp, aB);
        hbox_kernel<<<B_ * L_ * H_, W_, 0, stream>>>(bB, tmp);
        vbox_kernel<<<dim3(W_ / 64, H_ / 32, B_ * L_), vblk, 0, stream>>>(tmp, bB);

        // Qf = ma.I + mb ; msg = mu @ Qf (WMMA) ; ql = cu - msg
        apply_mu_kernel<<<600, T, 0, stream>>>(img, aB, bB, cu, ql, mxf);
    }

    // --- expectation --------------------------------------------------------
    final_kernel<<<GP, T, 0, stream>>>(ql, mxf, out);
}